// AdaptivePatchEmbed_5128190951570
// MI455X (gfx1250) — compile-verified
//
#include <hip/hip_runtime.h>

typedef __attribute__((ext_vector_type(2))) float v2f;
typedef __attribute__((ext_vector_type(4))) float v4f;
typedef __attribute__((ext_vector_type(8))) float v8f;

#define DD   768
#define KKT  3072          // 2*2*768
#define N0T  1408
#define N1T  336
#define N2T  84
#define NTOT 1828          // 1408+336+84
#define BB   8
#define S_I  98304         // W*T*D = 32*4*768
#define S_J  3072          // T*D   = 4*768
#define NTIL (DD/16)       // 48 N-tiles

// ---------------------------------------------------------------------------
// Repack w[d][c][i][j] (D,D,2,2) into Wt[k/4][n][4] with k = i*1536 + j*768 + c,
// so a GEMM lane grabs its two K values with one b64 load, 16B stride across lanes.
// ---------------------------------------------------------------------------
__global__ void repack_w(const float* __restrict__ w1, const float* __restrict__ w2a,
                         const float* __restrict__ w2b, float* __restrict__ ws) {
  long idx = (long)blockIdx.x * blockDim.x + threadIdx.x;
  const long per = (long)KKT * DD;
  if (idx >= 3 * per) return;
  int wsel = (int)(idx / per);
  int e    = (int)(idx - (long)wsel * per);
  const float* w = (wsel == 0) ? w1 : ((wsel == 1) ? w2a : w2b);
  float* dst = ws + (long)wsel * per;
  int k = e / DD;
  int n = e - k * DD;
  int i  = (k >= 1536) ? 1 : 0;
  int km = k - i * 1536;
  int j  = (km >= DD) ? 1 : 0;
  int c  = km - j * DD;
  dst[(long)(k >> 2) * (DD * 4) + n * 4 + (k & 3)] = w[(long)n * KKT + c * 4 + i * 2 + j];
}

// ---------------------------------------------------------------------------
// tok0: pure gather, float4 per thread
// ---------------------------------------------------------------------------
__global__ void tok0_gather(const float* __restrict__ base, const int* __restrict__ d0,
                            float* __restrict__ out) {
  long idx = (long)blockIdx.x * blockDim.x + threadIdx.x;
  const long total = (long)BB * N0T * (DD / 4);
  if (idx >= total) return;
  int d4   = (int)(idx % (DD / 4));
  int rest = (int)(idx / (DD / 4));
  int n = rest % N0T;
  int b = rest / N0T;
  int y = d0[n * 3 + 0], x = d0[n * 3 + 1], t = d0[n * 3 + 2];
  long src = ((((long)b * 32 + y) * 32 + x) * 4 + t) * (long)DD + d4 * 4;
  v4f v = *(const v4f*)(base + src);
  *(v4f*)(out + ((long)b * NTOT + n) * DD + d4 * 4) = v;
}

// ---------------------------------------------------------------------------
// positions: (y, x, size, t) int32 rows
// ---------------------------------------------------------------------------
__global__ void fill_pos(const int* __restrict__ d0, const int* __restrict__ d1,
                         const int* __restrict__ d2, int* __restrict__ pos) {
  int idx = blockIdx.x * blockDim.x + threadIdx.x;
  if (idx >= NTOT) return;
  const int* d; int o, s;
  if (idx < N0T)             { d = d0; o = idx;              s = 1; }
  else if (idx < N0T + N1T)  { d = d1; o = idx - N0T;        s = 2; }
  else                       { d = d2; o = idx - N0T - N1T;  s = 4; }
  pos[idx * 4 + 0] = d[o * 3 + 0];
  pos[idx * 4 + 1] = d[o * 3 + 1];
  pos[idx * 4 + 2] = s;
  pos[idx * 4 + 3] = d[o * 3 + 2];
}

// ---------------------------------------------------------------------------
// Stage-1 GEMM: out_tok1[r, n] = sum_k base_gather(r, k) * W1t(k, n) + b1[n]
// One wave per 16x16 tile; K-loop step 4 on V_WMMA_F32_16X16X4_F32.
// A layout per ISA: lane<16 holds K={0,1}, lane>=16 holds K={2,3} (v0=even,v1=odd).
// ---------------------------------------------------------------------------
__global__ __launch_bounds__(32) void gemm_tok1(
    const float* __restrict__ base, const int* __restrict__ d1,
    const float* __restrict__ Wt, const float* __restrict__ bias,
    float* __restrict__ out) {
  const int tile = blockIdx.x;
  const int tn = tile % NTIL;
  const int tm = tile / NTIL;
  const int lane = (int)threadIdx.x;
  const int hf  = lane >> 4;
  const int l16 = lane & 15;
  const int n0  = tn * 16;

  const int r  = tm * 16 + l16;
  const int b  = r / N1T;
  const int nn = r - b * N1T;
  const int y  = d1[nn * 3 + 0];
  const int x  = d1[nn * 3 + 1];
  const int t  = d1[nn * 3 + 2];
  const long rowBase = ((((long)b * 32 + y) * 32 + x) * 4 + t) * (long)DD;

  const float* wp = Wt + (n0 + l16) * 4 + 2 * hf;

  v8f acc = {0.f, 0.f, 0.f, 0.f, 0.f, 0.f, 0.f, 0.f};
#pragma unroll 4
  for (int k = 0; k < KKT; k += 4) {
    int i  = (k >= 1536) ? 1 : 0;
    int km = k - i * 1536;
    int j  = (km >= DD) ? 1 : 0;
    int c  = km - j * DD;
    v2f a   = *(const v2f*)(base + rowBase + i * S_I + j * S_J + c + 2 * hf);
    v2f bfr = *(const v2f*)(wp + (long)(k >> 2) * (DD * 4));
    acc = __builtin_amdgcn_wmma_f32_16x16x4_f32(false, a, false, bfr, (short)0, acc,
                                                false, false);
  }
  const float bv = bias[n0 + l16];
#pragma unroll
  for (int v = 0; v < 8; ++v) {
    int row = tm * 16 + v + 8 * hf;          // C/D layout: VGPR v -> M = v + 8*half
    int b2  = row / N1T;
    int rn  = row - b2 * N1T;
    out[((long)b2 * NTOT + N0T + rn) * DD + n0 + l16] = acc[v] + bv;
  }
}

// ---------------------------------------------------------------------------
// Stage-2a GEMM: rows are (q, h, w) quadrants of 4x4 blocks -> Y[q][h][w][d]
// ---------------------------------------------------------------------------
__global__ __launch_bounds__(32) void gemm_tok2a(
    const float* __restrict__ base, const int* __restrict__ d2,
    const float* __restrict__ Wt, const float* __restrict__ bias,
    float* __restrict__ Y) {
  const int tile = blockIdx.x;
  const int tn = tile % NTIL;
  const int tm = tile / NTIL;
  const int lane = (int)threadIdx.x;
  const int hf  = lane >> 4;
  const int l16 = lane & 15;
  const int n0  = tn * 16;

  const int r  = tm * 16 + l16;              // 0..2687
  const int q  = r >> 2;
  const int hw = r & 3;
  const int b  = q / N2T;
  const int nn = q - b * N2T;
  const int y  = d2[nn * 3 + 0] + 2 * (hw >> 1);
  const int x  = d2[nn * 3 + 1] + 2 * (hw & 1);
  const int t  = d2[nn * 3 + 2];
  const long rowBase = ((((long)b * 32 + y) * 32 + x) * 4 + t) * (long)DD;

  const float* wp = Wt + (n0 + l16) * 4 + 2 * hf;

  v8f acc = {0.f, 0.f, 0.f, 0.f, 0.f, 0.f, 0.f, 0.f};
#pragma unroll 4
  for (int k = 0; k < KKT; k += 4) {
    int i  = (k >= 1536) ? 1 : 0;
    int km = k - i * 1536;
    int j  = (km >= DD) ? 1 : 0;
    int c  = km - j * DD;
    v2f a   = *(const v2f*)(base + rowBase + i * S_I + j * S_J + c + 2 * hf);
    v2f bfr = *(const v2f*)(wp + (long)(k >> 2) * (DD * 4));
    acc = __builtin_amdgcn_wmma_f32_16x16x4_f32(false, a, false, bfr, (short)0, acc,
                                                false, false);
  }
  const float bv = bias[n0 + l16];
#pragma unroll
  for (int v = 0; v < 8; ++v) {
    int row = tm * 16 + v + 8 * hf;
    int q2  = row >> 2;
    int hw2 = row & 3;                        // hw2*768 == i*1536 + j*768 for stage-b K order
    Y[(long)q2 * KKT + hw2 * DD + n0 + l16] = acc[v] + bv;
  }
}

// ---------------------------------------------------------------------------
// Stage-2b GEMM: contiguous A rows from Y (q-major, K = (i,j,c) flat)
// ---------------------------------------------------------------------------
__global__ __launch_bounds__(32) void gemm_tok2b(
    const float* __restrict__ Y, const float* __restrict__ Wt,
    const float* __restrict__ bias, float* __restrict__ out) {
  const int tile = blockIdx.x;
  const int tn = tile % NTIL;
  const int tm = tile / NTIL;
  const int lane = (int)threadIdx.x;
  const int hf  = lane >> 4;
  const int l16 = lane & 15;
  const int n0  = tn * 16;

  const int r = tm * 16 + l16;               // q index, 0..671
  const float* arow = Y + (long)r * KKT + 2 * hf;
  const float* wp   = Wt + (n0 + l16) * 4 + 2 * hf;

  v8f acc = {0.f, 0.f, 0.f, 0.f, 0.f, 0.f, 0.f, 0.f};
#pragma unroll 4
  for (int k = 0; k < KKT; k += 4) {
    v2f a   = *(const v2f*)(arow + k);
    v2f bfr = *(const v2f*)(wp + (long)(k >> 2) * (DD * 4));
    acc = __builtin_amdgcn_wmma_f32_16x16x4_f32(false, a, false, bfr, (short)0, acc,
                                                false, false);
  }
  const float bv = bias[n0 + l16];
#pragma unroll
  for (int v = 0; v < 8; ++v) {
    int row = tm * 16 + v + 8 * hf;
    int b2  = row / N2T;
    int rn  = row - b2 * N2T;
    out[((long)b2 * NTOT + (N0T + N1T) + rn) * DD + n0 + l16] = acc[v] + bv;
  }
}

// ---------------------------------------------------------------------------
extern "C" void kernel_launch(void* const* d_in, const int* in_sizes, int n_in,
                              void* d_out, int out_size, void* d_ws, size_t ws_size,
                              hipStream_t stream) {
  const float* base = (const float*)d_in[0];
  const int*   d0   = (const int*)d_in[1];
  const int*   d1   = (const int*)d_in[2];
  const int*   d2   = (const int*)d_in[3];
  const float* w1   = (const float*)d_in[4];
  const float* b1   = (const float*)d_in[5];
  const float* w2a  = (const float*)d_in[6];
  const float* b2a  = (const float*)d_in[7];
  const float* w2b  = (const float*)d_in[8];
  const float* b2b  = (const float*)d_in[9];

  float* ws   = (float*)d_ws;
  const long per = (long)KKT * DD;           // 2,359,296 floats per repacked weight
  float* Wt1  = ws;
  float* Wt2a = ws + per;
  float* Wt2b = ws + 2 * per;
  float* Y2a  = ws + 3 * per;                // 672*3072 floats; total ws ~36.6 MB

  float* tokens = (float*)d_out;
  int*   pos    = (int*)d_out + (long)BB * NTOT * DD;

  {
    long n = 3 * per;
    repack_w<<<(int)((n + 255) / 256), 256, 0, stream>>>(w1, w2a, w2b, ws);
  }
  {
    long n = (long)BB * N0T * (DD / 4);
    tok0_gather<<<(int)((n + 255) / 256), 256, 0, stream>>>(base, d0, tokens);
  }
  fill_pos<<<(NTOT + 255) / 256, 256, 0, stream>>>(d0, d1, d2, pos);

  gemm_tok1 <<<(BB * N1T / 16) * NTIL,     32, 0, stream>>>(base, d1, Wt1,  b1,  tokens); // 8064 waves
  gemm_tok2a<<<(BB * N2T * 4 / 16) * NTIL, 32, 0, stream>>>(base, d2, Wt2a, b2a, Y2a);    // 8064 waves
  gemm_tok2b<<<(BB * N2T / 16) * NTIL,     32, 0, stream>>>(Y2a, Wt2b, b2b, tokens);      // 2016 waves
}